// AbstractReluConv_79422535238274
// MI455X (gfx1250) — compile-verified
//
#include <hip/hip_runtime.h>

// DeepPoly ReLU relaxation, elementwise over (x, low, high) -> (x', low', high').
// Pure streaming kernel: 805 MB of traffic, ~34.6 us roofline at 23.3 TB/s.
// Strategy: 128-bit global loads/stores with non-temporal cache hints (gfx1250
// TH_NT), wave32-friendly 256-thread blocks, one float4 per thread.

typedef float v4f __attribute__((ext_vector_type(4)));

__device__ __forceinline__ void relax1(float x, float l, float h,
                                       float& xo, float& lo_o, float& hi_o) {
    const bool crossing = (l < 0.0f) && (h > 0.0f);
    const bool nonpos   = (h <= 0.0f);

    // Safe denominator: only meaningful where crossing (there h - l > 0).
    const float denom = crossing ? (h - l) : 1.0f;
    const float inv   = 1.0f / denom;

    const float ub_slope   = h * inv;
    const float ub_int     = -(l * h) * inv;
    const float high_cross = ub_slope * h + ub_int;

    const float lam       = (l * l > h * h) ? 0.0f : 1.0f;
    const float low_cross = lam * l;

    hi_o = crossing ? high_cross : (nonpos ? 0.0f : h);
    lo_o = crossing ? low_cross  : (nonpos ? 0.0f : l);
    xo   = fmaxf(x, 0.0f);
}

__global__ void __launch_bounds__(256)
deeppoly_relu_v4(const float* __restrict__ x,
                 const float* __restrict__ lo,
                 const float* __restrict__ hi,
                 float* __restrict__ out,
                 int n, int nv) {
    float* __restrict__ x_out  = out;
    float* __restrict__ lo_out = out + (size_t)n;
    float* __restrict__ hi_out = out + 2 * (size_t)n;

    const int tid    = blockIdx.x * blockDim.x + threadIdx.x;
    const int stride = gridDim.x * blockDim.x;

    const v4f* __restrict__ xv4 = (const v4f*)x;
    const v4f* __restrict__ lv4 = (const v4f*)lo;
    const v4f* __restrict__ hv4 = (const v4f*)hi;
    v4f* __restrict__ xo4 = (v4f*)x_out;
    v4f* __restrict__ lo4 = (v4f*)lo_out;
    v4f* __restrict__ ho4 = (v4f*)hi_out;

    for (int i = tid; i < nv; i += stride) {
        // Streaming data, touched exactly once: non-temporal 128-bit accesses
        // (global_load_b128 / global_store_b128 with th:NT on gfx1250).
        v4f xv = __builtin_nontemporal_load(xv4 + i);
        v4f lv = __builtin_nontemporal_load(lv4 + i);
        v4f hv = __builtin_nontemporal_load(hv4 + i);

        v4f xr, lr, hr;
#pragma unroll
        for (int c = 0; c < 4; ++c) {
            float a = xv[c], l = lv[c], h = hv[c];
            float xo, lo_o, hi_o;
            relax1(a, l, h, xo, lo_o, hi_o);
            xr[c] = xo;
            lr[c] = lo_o;
            hr[c] = hi_o;
        }

        __builtin_nontemporal_store(xr, xo4 + i);
        __builtin_nontemporal_store(lr, lo4 + i);
        __builtin_nontemporal_store(hr, ho4 + i);
    }

    // Scalar tail (N % 4 != 0 case; N here is divisible by 4, kept general).
    const int tail = n - (nv << 2);
    if (tid < tail) {
        const int i = (nv << 2) + tid;
        float xo, lo_o, hi_o;
        relax1(x[i], lo[i], hi[i], xo, lo_o, hi_o);
        x_out[i]  = xo;
        lo_out[i] = lo_o;
        hi_out[i] = hi_o;
    }
}

extern "C" void kernel_launch(void* const* d_in, const int* in_sizes, int n_in,
                              void* d_out, int out_size, void* d_ws, size_t ws_size,
                              hipStream_t stream) {
    (void)n_in; (void)out_size; (void)d_ws; (void)ws_size;

    const float* x  = (const float*)d_in[0];
    const float* lo = (const float*)d_in[1];
    const float* hi = (const float*)d_in[2];
    float* out = (float*)d_out;

    const int n  = in_sizes[0];       // 8*2048*2048 = 33,554,432
    const int nv = n >> 2;            // float4 vectors per array

    const int threads = 256;          // 8 wave32 waves per block
    int blocks = (nv + threads - 1) / threads;
    if (blocks < 1) blocks = 1;

    deeppoly_relu_v4<<<blocks, threads, 0, stream>>>(x, lo, hi, out, n, nv);
}